// TreeLSTMCell_63831803953679
// MI455X (gfx1250) — compile-verified
//
#include <hip/hip_runtime.h>
#include <hip/hip_bf16.h>

typedef __attribute__((ext_vector_type(2))) float v2f;
typedef __attribute__((ext_vector_type(8))) float v8f;

#define HID 256
#define INSZ 768
#define GATES4 1024
#define BN 128          // output columns per wave = 8 chains of 16
#define BM 32           // output rows per wave = 2 row-tiles of 16

// C[M, coff+0..coff+255] (+)= A[M,K] @ Bz[256,K]^T, Bz/coff selected by blockIdx.z.
// LDB is a compile-time template so all 8 B column-group addresses collapse to one
// base pointer + immediate offsets (chain stride = 16*LDB*4 bytes).
// One wave32 per 32x128 C tile: 16 independent v_wmma_f32_16x16x4_f32 chains;
// every B fragment is reused by both row-tiles (10 loads / 16 WMMAs per k-step).
// __launch_bounds__(32,1): single-wave block, allow full VGPR budget (no spills).
// Fragment layouts per CDNA5 ISA 7.12.2:
//   A 16x4:  lanes0-15 hold (M=lane, K=0..1); lanes16-31 hold K=2..3
//   B 4x16:  mirrored, N = lane%16, K pair = 2*(lane/16)
//   D 16x16: VGPR r -> row 8*(lane/16)+r, col = lane%16
template <int LDB, int K>
__global__ __launch_bounds__(32, 1)
void gemm_nt_wmma(float* __restrict__ C,
                  const float* __restrict__ A,
                  const float* __restrict__ B0,
                  const float* __restrict__ B1,
                  const float* __restrict__ B2,
                  const float* __restrict__ B3,
                  int o0, int o1, int o2, int o3,
                  int M, int lda, int ldc, int accumulate) {
    const int lane = threadIdx.x;          // blockDim.x == 32
    const int hl   = lane >> 4;            // 0 or 1
    const int idx  = lane & 15;
    const int tn   = blockIdx.x;           // over 256/BN == 2 (N is always 256)
    const int tm   = blockIdx.y;           // over ceil(M/BM)

    // Select weight matrix + C column offset per gate (uniform scalar branch).
    const float* __restrict__ B;
    int coff;
    switch (blockIdx.z) {
        case 0:  B = B0; coff = o0; break;
        case 1:  B = B1; coff = o1; break;
        case 2:  B = B2; coff = o2; break;
        default: B = B3; coff = o3; break;
    }

    int mrow0 = tm * BM + idx;      if (mrow0 >= M) mrow0 = M - 1;  // clamped rows never stored
    int mrow1 = tm * BM + 16 + idx; if (mrow1 >= M) mrow1 = M - 1;
    const float* __restrict__ Ap0 = A + (size_t)mrow0 * lda + 2 * hl;
    const float* __restrict__ Ap1 = A + (size_t)mrow1 * lda + 2 * hl;
    // One base pointer; chain g lives at constant offset g*16*LDB floats.
    const float* __restrict__ Bp = B + (size_t)(tn * BN + idx) * LDB + 2 * hl;

    v8f acc0[8] = {};
    v8f acc1[8] = {};
    #pragma unroll 1
    for (int k = 0; k < K; k += 4) {
        v2f a0, a1, b[8];
        a0.x = Ap0[k]; a0.y = Ap0[k + 1];
        a1.x = Ap1[k]; a1.y = Ap1[k + 1];
        #pragma unroll
        for (int g = 0; g < 8; ++g) {
            b[g].x = Bp[g * 16 * LDB + k];
            b[g].y = Bp[g * 16 * LDB + k + 1];
        }
        #pragma unroll
        for (int g = 0; g < 8; ++g) {
            acc0[g] = __builtin_amdgcn_wmma_f32_16x16x4_f32(
                false, a0, false, b[g], (short)0, acc0[g], false, false);
            acc1[g] = __builtin_amdgcn_wmma_f32_16x16x4_f32(
                false, a1, false, b[g], (short)0, acc1[g], false, false);
        }
    }

    #pragma unroll
    for (int g = 0; g < 8; ++g) {
        const int col = coff + tn * BN + g * 16 + idx;
        #pragma unroll
        for (int r = 0; r < 8; ++r) {
            const int row0 = tm * BM + 8 * hl + r;
            if (row0 < M) {
                const size_t off = (size_t)row0 * ldc + col;
                float v = acc0[g][r];
                if (accumulate) v += C[off];
                C[off] = v;
            }
            const int row1 = tm * BM + 16 + 8 * hl + r;
            if (row1 < M) {
                const size_t off = (size_t)row1 * ldc + col;
                float v = acc1[g][r];
                if (accumulate) v += C[off];
                C[off] = v;
            }
        }
    }
}

// sum_h[j,g] = sum over 8 children of h_child[8j+k, g]
__global__ void child_sum_kernel(float* __restrict__ sum_h,
                                 const float* __restrict__ h_child,
                                 int nParent) {
    const int t = blockIdx.x * blockDim.x + threadIdx.x;
    if (t >= nParent * HID) return;
    const int j = t >> 8, g = t & (HID - 1);
    const float* p = h_child + ((size_t)j * 8) * HID + g;
    float s = 0.f;
    #pragma unroll
    for (int k = 0; k < 8; ++k) s += p[(size_t)k * HID];
    sum_h[t] = s;
}

__device__ __forceinline__ float sigmoidf_(float x) {
    return 1.0f / (1.0f + __expf(-x));        // x->-inf: 0;  x->+inf: 1
}
__device__ __forceinline__ float tanhf_(float x) {
    return 1.0f - 2.0f / (__expf(2.0f * x) + 1.0f);   // saturation-safe both ends
}

// Fused TreeLSTM cell activation for one level.
// pre[j, 0:256)=i, [256:512)=f, [512:768)=o, [768:1024)=u (pre-activation, no bias)
// fh[child, g] = (h_child @ U_f^T), c_child = child cell states.
__global__ void tree_act_kernel(float* __restrict__ h, float* __restrict__ c,
                                const float* __restrict__ pre,
                                const float* __restrict__ fh,
                                const float* __restrict__ c_child,
                                const float* __restrict__ b_i,
                                const float* __restrict__ b_f,
                                const float* __restrict__ b_o,
                                const float* __restrict__ b_u,
                                int n, int isLeaf) {
    const int t = blockIdx.x * blockDim.x + threadIdx.x;
    if (t >= n * HID) return;
    const int j = t >> 8, g = t & (HID - 1);
    const float* row = pre + (size_t)j * GATES4;

    const float xi = row[g]            + b_i[g];
    const float xf = row[HID + g]      + b_f[g];
    const float xo = row[2 * HID + g]  + b_o[g];
    const float xu = row[3 * HID + g]  + b_u[g];

    const float ig = sigmoidf_(xi);
    const float og = sigmoidf_(xo);
    const float ug = tanhf_(xu);

    float sum_f = 0.f;
    if (!isLeaf) {
        const size_t base = ((size_t)j * 8) * HID + g;
        #pragma unroll
        for (int k = 0; k < 8; ++k) {
            const size_t ci = base + (size_t)k * HID;
            sum_f += sigmoidf_(xf + fh[ci]) * c_child[ci];
        }
    }
    const float cc = sum_f + ig * ug;
    c[t] = cc;
    h[t] = og * tanhf_(cc);
}

__global__ void copy_out_kernel(float* __restrict__ out,
                                const float* __restrict__ h_root,
                                const float* __restrict__ c_root) {
    const int t = blockIdx.x * blockDim.x + threadIdx.x;
    if (t < HID)            out[t] = h_root[t];
    else if (t < 2 * HID)   out[t] = c_root[t - HID];
}

extern "C" void kernel_launch(void* const* d_in, const int* in_sizes, int n_in,
                              void* d_out, int out_size, void* d_ws, size_t ws_size,
                              hipStream_t stream) {
    (void)in_sizes; (void)n_in; (void)out_size; (void)ws_size;

    const float* X   = (const float*)d_in[0];
    const float* W_i = (const float*)d_in[1];
    const float* b_i = (const float*)d_in[2];
    const float* U_i = (const float*)d_in[3];
    const float* W_f = (const float*)d_in[4];
    const float* b_f = (const float*)d_in[5];
    const float* U_f = (const float*)d_in[6];
    const float* W_o = (const float*)d_in[7];
    const float* b_o = (const float*)d_in[8];
    const float* U_o = (const float*)d_in[9];
    const float* W_u = (const float*)d_in[10];
    const float* b_u = (const float*)d_in[11];
    const float* U_u = (const float*)d_in[12];
    float* out = (float*)d_out;

    // Tree geometry (BRANCH=8, DEPTH=5), level-major storage.
    const int  sizes[6] = {1, 8, 64, 512, 4096, 32768};
    const long offs[6]  = {0, 1, 9, 73, 585, 4681};
    const long N_NODES  = 37449;
    const int  MAXLEV   = 32768;

    // Workspace layout (floats)
    float* ws    = (float*)d_ws;
    float* h_all = ws;                                   // N_NODES*256
    float* c_all = h_all + N_NODES * HID;                // N_NODES*256
    float* pre   = c_all + N_NODES * HID;                // MAXLEV*1024
    float* fh    = pre + (size_t)MAXLEV * GATES4;        // MAXLEV*256
    float* sum_h = fh + (size_t)MAXLEV * HID;            // (MAXLEV/8)*256

    for (int d = 5; d >= 0; --d) {
        const int n = sizes[d];
        const float* Xlev = X + offs[d] * INSZ;
        float* h_lev = h_all + offs[d] * HID;
        float* c_lev = c_all + offs[d] * HID;

        const int mt = (n + BM - 1) / BM;

        // pre[:, gate*256 ...] = X @ W_gate^T   (K=768), all 4 gates via grid.z
        gemm_nt_wmma<INSZ, INSZ><<<dim3(HID / BN, mt, 4), 32, 0, stream>>>(
            pre, Xlev, W_i, W_f, W_o, W_u,
            0, HID, 2 * HID, 3 * HID, n, INSZ, GATES4, 0);

        const int isLeaf = (d == 5);
        const float* c_child = c_all;  // dummy for leaf (unused when isLeaf)
        if (!isLeaf) {
            const int nc = n * 8;
            const float* h_child = h_all + offs[d + 1] * HID;
            c_child = c_all + offs[d + 1] * HID;

            // fh = h_child @ U_f^T   (per-child forget recurrence, batched over child level)
            gemm_nt_wmma<HID, HID><<<dim3(HID / BN, (nc + BM - 1) / BM, 1), 32, 0, stream>>>(
                fh, h_child, U_f, U_f, U_f, U_f,
                0, 0, 0, 0, nc, HID, HID, 0);

            // sum_h = sum of 8 children h
            child_sum_kernel<<<(n * HID + 255) / 256, 256, 0, stream>>>(sum_h, h_child, n);

            // pre[:, i/o/u] += sum_h @ U^T   (3 gates via grid.z)
            gemm_nt_wmma<HID, HID><<<dim3(HID / BN, mt, 3), 32, 0, stream>>>(
                pre, sum_h, U_i, U_o, U_u, U_u,
                0, 2 * HID, 3 * HID, 3 * HID, n, HID, GATES4, 1);
        }

        tree_act_kernel<<<(n * HID + 255) / 256, 256, 0, stream>>>(
            h_lev, c_lev, pre, fh, c_child, b_i, b_f, b_o, b_u, n, isLeaf);
    }

    // d_out = concat(h[0:1], c[0:1]) = 512 floats
    copy_out_kernel<<<2, 256, 0, stream>>>(out, h_all, c_all);
}